// GNNModel_72189810311316
// MI455X (gfx1250) — compile-verified
//
#include <hip/hip_runtime.h>

#define NNODE 25000
#define NEDGE 500000
#define RWORDS (NNODE * 64)

typedef __attribute__((ext_vector_type(2))) float v2f;
typedef __attribute__((ext_vector_type(8))) float v8f;

#define ACT_NONE 0
#define ACT_RELU 1
#define ACT_SIG  2

// ---- CDNA5 async global->LDS staging (ASYNCcnt path), with safe fallback ----
#if defined(__HIP_DEVICE_COMPILE__) && __has_builtin(__builtin_amdgcn_global_load_async_to_lds_b128)
#define HAS_ASYNC_LDS 1
#else
#define HAS_ASYNC_LDS 0
#endif

typedef int vi4 __attribute__((__vector_size__(4 * sizeof(int))));

__device__ __forceinline__ void copy16_g2l(const float* g, float* l) {
#if HAS_ASYNC_LDS
    __builtin_amdgcn_global_load_async_to_lds_b128(
        (__attribute__((address_space(1))) vi4*)(void*)g,
        (__attribute__((address_space(3))) vi4*)(void*)l, 0, 0);
#else
    *(float4*)l = *(const float4*)g;
#endif
}

__device__ __forceinline__ void wait_lds_copies() {
#if HAS_ASYNC_LDS
#if __has_builtin(__builtin_amdgcn_s_wait_asynccnt)
    __builtin_amdgcn_s_wait_asynccnt(0);
#else
    asm volatile("s_wait_asynccnt 0" ::: "memory");
#endif
#endif
}

__device__ __forceinline__ float act_apply(float v, int act) {
    if (act == ACT_RELU) v = fmaxf(v, 0.0f);
    else if (act == ACT_SIG) v = 1.0f / (1.0f + __expf(-v));
    return v;
}

// Y[N,64] = act( [X0*s0 | X1][N,K] @ W[K,64] + b ), K = 64 or 128.
// 256 threads = 8 waves; wave w -> (rowtile w>>2, coltile w&3) of a 32x64 block.
__global__ __launch_bounds__(256)
void gemm_n64(const float* __restrict__ X0, const float* __restrict__ X1,
              const float* __restrict__ s0p, const float* __restrict__ W,
              const float* __restrict__ bias, int K, int act, int N,
              float* __restrict__ Y)
{
    __shared__ float Xs[32][129];                        // padded -> conflict-free
    __shared__ __attribute__((aligned(16))) float Ws[128][64];
    __shared__ float Bs[64];
    const int tid = threadIdx.x;
    const int r0  = blockIdx.x * 32;
    const float s0 = s0p ? s0p[0] : 1.0f;
    const int ksh = (K >> 7) + 6;       // 64 -> 6, 128 -> 7
    const int kmask = K - 1;

    // async-stage W (K*64 floats = K*16 16B chunks) while we transform X below
    for (int idx = tid; idx < (K << 4); idx += 256)
        copy16_g2l(W + idx * 4, &Ws[0][0] + idx * 4);
    if (tid < 64) Bs[tid] = bias ? bias[tid] : 0.0f;
    for (int idx = tid; idx < (K << 5); idx += 256) {
        int r = idx >> ksh, k = idx & kmask;
        int row = r0 + r;
        float v = 0.0f;
        if (row < N)
            v = (k < 64) ? X0[row * 64 + k] * s0 : X1[row * 64 + (k - 64)];
        Xs[r][k] = v;
    }
    wait_lds_copies();
    __syncthreads();

    const int wave = tid >> 5;
    const int lane = tid & 31;
    const int rt = wave >> 2;
    const int ct = wave & 3;
    const int l  = lane & 15;
    const int hi = lane >> 4;           // 0: K 0..1, 1: K 2..3 within a k-step

    v8f acc = {};
    for (int kk = 0; kk < K; kk += 4) {
        v2f a, b;
        a.x = Xs[rt * 16 + l][kk + 2 * hi + 0];
        a.y = Xs[rt * 16 + l][kk + 2 * hi + 1];
        b.x = Ws[kk + 2 * hi + 0][ct * 16 + l];
        b.y = Ws[kk + 2 * hi + 1][ct * 16 + l];
        acc = __builtin_amdgcn_wmma_f32_16x16x4_f32(false, a, false, b,
                                                    (short)0, acc, false, false);
    }

    const int col = ct * 16 + l;
    const float bc = Bs[col];
#pragma unroll
    for (int r = 0; r < 8; ++r) {
        int row = r0 + rt * 16 + hi * 8 + r;   // C/D: lanes 0-15 M=r, 16-31 M=8+r
        float v = act_apply(acc[r] + bc, act);
        if (row < N) Y[row * 64 + col] = v;
    }
}

// Fused edge message + GEMM(W_final) + scatter-add into agg[i].
// 128 threads (4 waves: one per 16-col tile); 16 chunks of 16 edges per block.
__global__ __launch_bounds__(128)
void conv_edge(const float* __restrict__ LT, const float* __restrict__ RT,
               const float* __restrict__ eattr, const float* __restrict__ eshift,
               const float* __restrict__ escale, const float* __restrict__ edgeW,
               const float* __restrict__ fscale, const float* __restrict__ finalW,
               const float* __restrict__ finalb,
               const int* __restrict__ jIdx, const int* __restrict__ iIdx,
               int E, float* __restrict__ agg)
{
    __shared__ __attribute__((aligned(16))) float Wf[64][64];
    __shared__ float Ms[16][65];
    __shared__ float We[64];
    __shared__ float Bs[64];
    __shared__ int   iS[16];
    __shared__ int   jS[16];
    __shared__ float efS[16];

    const int tid = threadIdx.x;
    for (int idx = tid; idx < 1024; idx += 128)          // 16KB async to LDS
        copy16_g2l(finalW + idx * 4, &Wf[0][0] + idx * 4);
    if (tid < 64) { We[tid] = edgeW[tid]; Bs[tid] = finalb[tid]; }
    const float esh = eshift[0], esc = escale[0], fs = fscale[0];
    wait_lds_copies();

    const int wave = tid >> 5;     // col tile 0..3
    const int lane = tid & 31;
    const int l  = lane & 15;
    const int hi = lane >> 4;

    for (int c = 0; c < 16; ++c) {
        const int e0 = (blockIdx.x * 16 + c) * 16;
        if (e0 >= E) break;                    // uniform across block
        __syncthreads();                       // previous chunk fully done
        if (tid < 16) {
            int eg = e0 + tid;
            int ok = eg < E;
            iS[tid]  = ok ? iIdx[eg] : 0;
            jS[tid]  = ok ? jIdx[eg] : 0;
            efS[tid] = ok ? (eattr[eg] + esh) * esc : 0.0f;
        }
        __syncthreads();
        for (int idx = tid; idx < 1024; idx += 128) {
            int e = idx >> 6, col = idx & 63;
            float v = 0.0f;
            if (e0 + e < E) {
                v = LT[jS[e] * 64 + col] + efS[e] * We[col] + RT[iS[e] * 64 + col];
                v = fmaxf(v * fs, 0.0f);
            }
            Ms[e][col] = v;
        }
        __syncthreads();

        v8f acc = {};
        for (int kk = 0; kk < 64; kk += 4) {
            v2f a, b;
            a.x = Ms[l][kk + 2 * hi + 0];
            a.y = Ms[l][kk + 2 * hi + 1];
            b.x = Wf[kk + 2 * hi + 0][wave * 16 + l];
            b.y = Wf[kk + 2 * hi + 1][wave * 16 + l];
            acc = __builtin_amdgcn_wmma_f32_16x16x4_f32(false, a, false, b,
                                                        (short)0, acc, false, false);
        }
        const int col = wave * 16 + l;
        const float bc = Bs[col];
#pragma unroll
        for (int r = 0; r < 8; ++r) {
            int e = hi * 8 + r;
            if (e0 + e < E)
                atomicAdd(&agg[iS[e] * 64 + col], acc[r] + bc);
        }
    }
}

// G[dst[e]] += attr[e] * M[src[e]]  (64-wide rows, coalesced per edge)
__global__ __launch_bounds__(256)
void gcn_scatter(const float* __restrict__ attr, const float* __restrict__ M,
                 const int* __restrict__ src, const int* __restrict__ dst,
                 int E, float* __restrict__ G)
{
    int idx = blockIdx.x * 256 + threadIdx.x;
    int e = idx >> 6, col = idx & 63;
    if (e >= E) return;
    atomicAdd(&G[dst[e] * 64 + col], attr[e] * M[src[e] * 64 + col]);
}

// Wave-per-row LayerNorm of (X + Z) over 64 features (wave32: 2 elems/lane).
__global__ __launch_bounds__(256)
void ln_kernel(const float* __restrict__ X, const float* __restrict__ Z,
               const float* __restrict__ g, const float* __restrict__ b,
               int N, float* __restrict__ Y)
{
    int wave = threadIdx.x >> 5, lane = threadIdx.x & 31;
    int row = blockIdx.x * 8 + wave;
    if (row >= N) return;
    float x0 = X[row * 64 + lane]      + Z[row * 64 + lane];
    float x1 = X[row * 64 + 32 + lane] + Z[row * 64 + 32 + lane];
    float s = x0 + x1;
    for (int off = 16; off > 0; off >>= 1) s += __shfl_xor(s, off, 32);
    float m = s * (1.0f / 64.0f);
    float d0 = x0 - m, d1 = x1 - m;
    float v = d0 * d0 + d1 * d1;
    for (int off = 16; off > 0; off >>= 1) v += __shfl_xor(v, off, 32);
    float inv = rsqrtf(v * (1.0f / 64.0f) + 1e-5f);
    Y[row * 64 + lane]      = d0 * inv * g[lane]      + b[lane];
    Y[row * 64 + 32 + lane] = d1 * inv * g[32 + lane] + b[32 + lane];
}

// First embedding layer: din in {4,13}; Y = relu(((x+shift)*scale) @ W + b)
__global__ __launch_bounds__(256)
void embed_l1(const float* __restrict__ X, const float* __restrict__ W,
              const float* __restrict__ bias, const float* __restrict__ scale,
              const float* __restrict__ shift, int din, int N, float* __restrict__ Y)
{
    int idx = blockIdx.x * 256 + threadIdx.x;
    int n = idx >> 6, c = idx & 63;
    if (n >= N) return;
    float acc = bias[c];
    for (int k = 0; k < din; ++k)
        acc += (X[n * din + k] + shift[k]) * scale[k] * W[k * 64 + c];
    Y[n * 64 + c] = fmaxf(acc, 0.0f);
}

__global__ __launch_bounds__(256)
void copy4_f32(const float4* __restrict__ X, float4* __restrict__ Y, int n4) {
    int i = blockIdx.x * 256 + threadIdx.x;
    if (i < n4) Y[i] = X[i];
}

__global__ __launch_bounds__(256)
void fill04_f32(float4* __restrict__ Y, int n4) {
    int i = blockIdx.x * 256 + threadIdx.x;
    if (i < n4) Y[i] = make_float4(0.f, 0.f, 0.f, 0.f);
}

// ---------------------------------------------------------------------------
// Param leaf order (jax pytree, dicts sorted by key):
//  base=4
//  cons_emb: 0 l1.W  1 l1.b  2 l2.W  3 l2.b  4 scale  5 shift
//  conv_cv @6, conv_vc @18, each:
//    0 edge.W 1 final.W 2 final.b 3 final_scale 4 left.W 5 left.b
//    6 o1.W 7 o1.b 8 o2.W 9 o2.b 10 post_scale 11 right.W
//  edge_emb: 30 scale 31 shift
//  gcn layer L @32+16L:
//    0 cm0.W 1 cm0.b 2 cm1.W 3 cm1.b 4 jc.W 5 jc.b 6 jv.W 7 jv.b
//    8 ln_c.b 9 ln_c.g 10 ln_v.b 11 ln_v.g 12 vm0.W 13 vm0.b 14 vm1.W 15 vm1.b
//  var_emb @112: 0 l1.W 1 l1.b 2 l2.W 3 l2.b 4 scale 5 shift
// ---------------------------------------------------------------------------

extern "C" void kernel_launch(void* const* d_in, const int* in_sizes, int n_in,
                              void* d_out, int out_size, void* d_ws, size_t ws_size,
                              hipStream_t stream)
{
    (void)in_sizes; (void)n_in; (void)out_size; (void)ws_size;

    const float* cons_x = (const float*)d_in[0];
    const int*   eidx   = (const int*)d_in[1];
    const float* eattr  = (const float*)d_in[2];
    const float* var_x  = (const float*)d_in[3];
    const float* const* P = (const float* const*)(d_in + 4);

    const int* row = eidx;          // edge_indices[0] = constraint ids
    const int* col = eidx + NEDGE;  // edge_indices[1] = variable ids

    float* ws = (float*)d_ws;
    const long R = RWORDS;
    float* t0   = ws + 0 * R;
    float* t1   = ws + 1 * R;
    float* t2   = ws + 2 * R;
    float* t3   = ws + 3 * R;
    float* cons = ws + 4 * R;
    float* var  = ws + 5 * R;
    float* tcA  = ws + 6 * R;
    float* tcB  = ws + 7 * R;
    float* tvA  = ws + 8 * R;
    float* tvB  = ws + 9 * R;

    const int GB  = (NNODE + 31) / 32;           // 782 gemm blocks
    const int EL  = (NNODE * 64) / 256;          // 6250 elementwise blocks
    const int EL4 = (RWORDS / 4 + 255) / 256;    // 1563 float4 blocks
    const int EB  = (NEDGE + 255) / 256;         // 1954 conv_edge blocks
    const int SB  = (NEDGE * 64) / 256;          // 125000 scatter blocks
    const int LB  = (NNODE + 7) / 8;             // 3125 layernorm blocks
    const int R4  = RWORDS / 4;

    // ---- embeddings ----
    embed_l1<<<EL, 256, 0, stream>>>(cons_x, P[0], P[1], P[4], P[5], 4, NNODE, t0);
    gemm_n64<<<GB, 256, 0, stream>>>(t0, nullptr, nullptr, P[2], P[3], 64, ACT_RELU, NNODE, cons);
    embed_l1<<<EL, 256, 0, stream>>>(var_x, P[112], P[113], P[116], P[117], 13, NNODE, t0);
    gemm_n64<<<GB, 256, 0, stream>>>(t0, nullptr, nullptr, P[114], P[115], 64, ACT_RELU, NNODE, var);

    const float* eg_scale = P[30];
    const float* eg_shift = P[31];

    // ---- conv v -> c (params conv_vc): left=var (j=col), right=cons (i=row) ----
    {
        const float* const* Q = P + 18;
        gemm_n64<<<GB, 256, 0, stream>>>(var,  nullptr, nullptr, Q[4], Q[5],  64, ACT_NONE, NNODE, t0); // LT
        gemm_n64<<<GB, 256, 0, stream>>>(cons, nullptr, nullptr, Q[11], nullptr, 64, ACT_NONE, NNODE, t1); // RT
        fill04_f32<<<EL4, 256, 0, stream>>>((float4*)t2, R4);
        conv_edge<<<EB, 128, 0, stream>>>(t0, t1, eattr, eg_shift, eg_scale,
                                          Q[0], Q[3], Q[1], Q[2], col, row, NEDGE, t2);
        gemm_n64<<<GB, 256, 0, stream>>>(t2, cons, Q[10], Q[6], Q[7], 128, ACT_RELU, NNODE, t0); // h
        gemm_n64<<<GB, 256, 0, stream>>>(t0, nullptr, nullptr, Q[8], Q[9], 64, ACT_NONE, NNODE, cons);
    }
    // ---- conv c -> v (params conv_cv): left=cons (j=row), right=var (i=col) ----
    {
        const float* const* Q = P + 6;
        gemm_n64<<<GB, 256, 0, stream>>>(cons, nullptr, nullptr, Q[4], Q[5],  64, ACT_NONE, NNODE, t0);
        gemm_n64<<<GB, 256, 0, stream>>>(var,  nullptr, nullptr, Q[11], nullptr, 64, ACT_NONE, NNODE, t1);
        fill04_f32<<<EL4, 256, 0, stream>>>((float4*)t2, R4);
        conv_edge<<<EB, 128, 0, stream>>>(t0, t1, eattr, eg_shift, eg_scale,
                                          Q[0], Q[3], Q[1], Q[2], row, col, NEDGE, t2);
        gemm_n64<<<GB, 256, 0, stream>>>(t2, var, Q[10], Q[6], Q[7], 128, ACT_RELU, NNODE, t0);
        gemm_n64<<<GB, 256, 0, stream>>>(t0, nullptr, nullptr, Q[8], Q[9], 64, ACT_NONE, NNODE, var);
    }

    // ---- GCN stack ----
    copy4_f32<<<EL4, 256, 0, stream>>>((const float4*)cons, (float4*)tcA, R4);
    copy4_f32<<<EL4, 256, 0, stream>>>((const float4*)var,  (float4*)tvA, R4);
    float* tc_cur = tcA; float* tc_nxt = tcB;
    float* tv_cur = tvA; float* tv_nxt = tvB;

    for (int lyr = 0; lyr < 5; ++lyr) {
        const float* const* Q = P + 32 + 16 * lyr;
        gemm_n64<<<GB, 256, 0, stream>>>(cons, nullptr, nullptr, Q[0],  Q[1],  64, ACT_RELU, NNODE, t0);
        gemm_n64<<<GB, 256, 0, stream>>>(t0,   nullptr, nullptr, Q[2],  Q[3],  64, ACT_RELU, NNODE, t1); // mc
        gemm_n64<<<GB, 256, 0, stream>>>(var,  nullptr, nullptr, Q[12], Q[13], 64, ACT_RELU, NNODE, t0);
        gemm_n64<<<GB, 256, 0, stream>>>(t0,   nullptr, nullptr, Q[14], Q[15], 64, ACT_RELU, NNODE, t2); // mv
        fill04_f32<<<EL4, 256, 0, stream>>>((float4*)t3, R4);
        gcn_scatter<<<SB, 256, 0, stream>>>(eattr, t2, col, row, NEDGE, t3);   // sum_e attr*mv[col] -> rows
        fill04_f32<<<EL4, 256, 0, stream>>>((float4*)t0, R4);
        gcn_scatter<<<SB, 256, 0, stream>>>(eattr, t1, row, col, NEDGE, t0);   // sum_e attr*mc[row] -> cols
        ln_kernel<<<LB, 256, 0, stream>>>(t3, cons, Q[9],  Q[8],  NNODE, cons); // zc = LN(gc + zc)
        ln_kernel<<<LB, 256, 0, stream>>>(t0, var,  Q[11], Q[10], NNODE, var);  // zv = LN(gv + zv)
        gemm_n64<<<GB, 256, 0, stream>>>(cons, tc_cur, nullptr, Q[4], Q[5], 128, ACT_SIG, NNODE, tc_nxt);
        gemm_n64<<<GB, 256, 0, stream>>>(var,  tv_cur, nullptr, Q[6], Q[7], 128, ACT_SIG, NNODE, tv_nxt);
        float* tmp = tc_cur; tc_cur = tc_nxt; tc_nxt = tmp;
        tmp = tv_cur; tv_cur = tv_nxt; tv_nxt = tmp;
    }

    copy4_f32<<<EL4, 256, 0, stream>>>((const float4*)tv_cur, (float4*)d_out, R4);
}